// MY_NCE_LOSS_50672024158589
// MI455X (gfx1250) — compile-verified
//
#include <hip/hip_runtime.h>
#include <hip/hip_bf16.h>
#include <math.h>

typedef __attribute__((ext_vector_type(2))) float v2f;
typedef __attribute__((ext_vector_type(8))) float v8f;
typedef int v4i __attribute__((vector_size(4 * sizeof(int))));   // matches builtin pointee

#define BATCH       256
#define DIM         256
#define NUM_TRUE    1024
#define NUM_SAMPLED 16384
#define LOG_RANGE   11.512935464920228f   // log(NUM_CLASSES + 1)

#define MT   16                     // M tile per block
#define NT   256                    // N tile per block (8 waves x 2 sub-tiles)
#define KC   16                     // K chunk staged in LDS per pipeline stage
#define NCH  (NUM_SAMPLED / NT)     // 64 N-chunks
#define NKC  (DIM / KC)             // 16 K chunks
#define LDP  (KC + 4)               // 20 floats: 80B rows, 16B aligned, bank-stride 4*5
#define BSZ  (8 * 32 * LDP)         // 5120 floats per buffer (8 wave-private regions)

#if defined(__has_builtin)
#if __has_builtin(__builtin_amdgcn_global_load_async_to_lds_b128)
#define HAVE_ASYNC_LDS 1
#endif
#endif

typedef __attribute__((address_space(1))) v4i gv4i_t;   // global int4
typedef __attribute__((address_space(3))) v4i lv4i_t;   // LDS int4

template <int OFF>   // byte offset applied to BOTH global and LDS sides (ISA 08 §4.4)
__device__ __forceinline__ void copy16_g2l(const float* g, float* l) {
#if defined(HAVE_ASYNC_LDS)
    __builtin_amdgcn_global_load_async_to_lds_b128((gv4i_t*)g, (lv4i_t*)l, OFF, 0);
#else
    *(float4*)((char*)l + OFF) = *(const float4*)((const char*)g + OFF);
#endif
}

// Token-laundering waits: the returned value (always its input) flows into the
// LDS address computation, creating a hard data dependency so the scheduler
// CANNOT hoist dependent LDS accesses above the wait (it demonstrably hoists
// loads across bare wait intrinsics and asm memory clobbers).
__device__ __forceinline__ int wait_async_le4_tok(int t) {
#if defined(HAVE_ASYNC_LDS)
    asm volatile("s_wait_asynccnt 0x4" : "+v"(t) :: "memory");
#endif
    return t;
}
__device__ __forceinline__ int wait_async_zero_tok(int t) {
#if defined(HAVE_ASYNC_LDS)
    asm volatile("s_wait_asynccnt 0x0" : "+v"(t) :: "memory");
#endif
    return t;
}
__device__ __forceinline__ int wait_dscnt_zero_tok(int t) {
#if defined(HAVE_ASYNC_LDS)
    asm volatile("s_wait_dscnt 0x0" : "+v"(t) :: "memory");
#endif
    return t;
}

__device__ __forceinline__ float neg_log_expected_count(int id) {
    float idf = (float)id;
    float p  = logf((idf + 2.0f) / (idf + 1.0f)) * (1.0f / LOG_RANGE);
    float ec = -expm1f((float)NUM_SAMPLED * log1pf(-p));
    return logf(ec);
}

__device__ __forceinline__ float sigmoid_xent(float l, float target) {
    return fmaxf(l, 0.0f) - l * target + log1pf(expf(-fabsf(l)));
}

#define WMMA_F32(A, B, C) \
    __builtin_amdgcn_wmma_f32_16x16x4_f32(false, (A), false, (B), (short)0, (C), false, false)

// ---------------------------------------------------------------------------
// Sampled logits: C = inputs @ w[sampled]^T via V_WMMA_F32_16X16X4_F32.
// Wave-autonomous async pipeline: each lane async-stages its OWN gathered row
// into a wave-private LDS region (no barriers in the K loop); double-buffered
// with partial-counter waits (s_wait_asynccnt 0x4). A-tile read from global.
// grid: (NCH, BATCH/MT), block 256 (8 waves; each wave owns two 16x16 tiles).
// ---------------------------------------------------------------------------
__global__ __launch_bounds__(256) void nce_sampled_kernel(
    const float* __restrict__ inp, const float* __restrict__ w,
    const float* __restrict__ bias, const int* __restrict__ sampled,
    float* __restrict__ part /* [BATCH][NCH] */)
{
    __shared__ float Bs[2 * BSZ];
    __shared__ float redw[8][16];

    const int tid    = threadIdx.x;
    const int lane   = tid & 31;
    const int wv     = tid >> 5;
    const int nch    = blockIdx.x;          // 0..NCH-1
    const int m_base = blockIdx.y * MT;
    const int n_base = nch * NT;
    const int hl     = lane & 15;
    const int koff   = (lane < 16) ? 0 : 2; // ISA A/B half-wave K offset

    // Each lane owns ONE gathered row of its wave-private B region.
    const int    sid  = sampled[n_base + wv * 32 + lane];
    const float* gsrc = w + (size_t)sid * DIM;
    const int    ldst = wv * (32 * LDP) + lane * LDP;   // float index in a buffer

    auto stage = [&](int ck, int buf, int tok) {        // 4 async b128 per lane
        const float* g = gsrc + ck * KC;
        float*       l = Bs + buf * BSZ + ldst + tok;   // tok pins order
        copy16_g2l<0>(g, l);  copy16_g2l<16>(g, l);
        copy16_g2l<32>(g, l); copy16_g2l<48>(g, l);
    };

    v8f c00 = {0,0,0,0,0,0,0,0}, c01 = {0,0,0,0,0,0,0,0};   // tile0 K-split accums
    v8f c10 = {0,0,0,0,0,0,0,0}, c11 = {0,0,0,0,0,0,0,0};   // tile1 K-split accums

    const float* aptr  = inp + (size_t)(m_base + hl) * DIM + koff;  // A from global (L1-hot)
    const int    brow0 = wv * (32 * LDP) + hl * LDP;
    const int    brow1 = brow0 + 16 * LDP;

    int tok = 0;
    stage(0, 0, tok);                        // 2-deep pipeline prologue
    stage(1, 1, tok);
    #pragma unroll
    for (int ck = 0; ck < NKC; ++ck) {
        const int buf = ck & 1;
        // Retire the OLDER stage (async loads complete in order, ISA 08 §4.1);
        // the newer 4 copies stay in flight.
        tok = (ck < NKC - 1) ? wait_async_le4_tok(tok) : wait_async_zero_tok(tok);

        const float* Bb = Bs + buf * BSZ + tok;          // data-dep on the wait
        const float* Ab = aptr + ck * KC;
        #pragma unroll
        for (int s = 0; s < KC / 4; ++s) {
            v2f a  = *(const v2f*)(Ab + 4 * s);
            v2f b0 = *(const v2f*)(Bb + brow0 + 4 * s + koff);
            v2f b1 = *(const v2f*)(Bb + brow1 + 4 * s + koff);
            if (s & 1) { c01 = WMMA_F32(a, b0, c01); c11 = WMMA_F32(a, b1, c11); }
            else       { c00 = WMMA_F32(a, b0, c00); c10 = WMMA_F32(a, b1, c10); }
        }
        if (ck + 2 < NKC) {
            // Async writes are unordered vs pending indexed reads: ensure this
            // wave's reads of `buf` are serviced before restaging it.
            tok = wait_dscnt_zero_tok(tok);
            stage(ck + 2, buf, tok);
        }
    }

    // ---- epilogue: merge K-split accumulators, apply bias/offset/xent -----
    const int   n0   = wv * 32 + hl;
    const int   n1   = n0 + 16;
    const int   sid0 = sampled[n_base + n0];
    const int   sid1 = sampled[n_base + n1];
    const float bb0  = bias[sid0], bb1 = bias[sid1];
    const float lec0 = neg_log_expected_count(sid0);
    const float lec1 = neg_log_expected_count(sid1);

    float acc[8];
    #pragma unroll
    for (int i = 0; i < 8; ++i) {
        float l0 = (c00[i] + c01[i]) + bb0 - lec0;
        float l1 = (c10[i] + c11[i]) + bb1 - lec1;
        acc[i] = sigmoid_xent(l0, 0.0f) + sigmoid_xent(l1, 0.0f);
    }
    #pragma unroll
    for (int msk = 1; msk < 16; msk <<= 1) {   // sum n within each half-wave
        #pragma unroll
        for (int i = 0; i < 8; ++i)
            acc[i] += __shfl_xor(acc[i], msk, 32);
    }
    if (lane == 0) {
        #pragma unroll
        for (int i = 0; i < 8; ++i) redw[wv][i] = acc[i];
    }
    if (lane == 16) {
        #pragma unroll
        for (int i = 0; i < 8; ++i) redw[wv][8 + i] = acc[i];
    }
    __syncthreads();
    if (tid < 16) {
        float s = 0.f;
        #pragma unroll
        for (int wvi = 0; wvi < 8; ++wvi) s += redw[wvi][tid];   // fixed order
        part[(size_t)(m_base + tid) * NCH + nch] = s;
    }
}

// ---------------------------------------------------------------------------
// True logits: per-batch gather-dots (6% of FLOPs). One block per batch row.
// ---------------------------------------------------------------------------
__global__ __launch_bounds__(256) void nce_true_kernel(
    const float* __restrict__ inp, const float* __restrict__ w,
    const float* __restrict__ bias, const int* __restrict__ labels,
    float* __restrict__ tpart /* [BATCH] */)
{
    const int b    = blockIdx.x;
    const int tid  = threadIdx.x;
    const int lane = tid & 31;
    const int wv   = tid >> 5;
    __shared__ float redw[8];

    float4 x0 = *(const float4*)(inp + (size_t)b * DIM + lane * 8);
    float4 x1 = *(const float4*)(inp + (size_t)b * DIM + lane * 8 + 4);

    const float tgt = 1.0f / (float)NUM_TRUE;
    float acc = 0.f;
    for (int t = wv; t < NUM_TRUE; t += 8) {
        const int lab = labels[(size_t)b * NUM_TRUE + t];
        const float* wr = w + (size_t)lab * DIM + lane * 8;
        int nt = t + 8;
        if (nt < NUM_TRUE) {
            int nlab = labels[(size_t)b * NUM_TRUE + nt];
            __builtin_prefetch(w + (size_t)nlab * DIM + lane * 8, 0, 0);
        }
        float4 w0 = *(const float4*)(wr);
        float4 w1 = *(const float4*)(wr + 4);
        float d = x0.x * w0.x + x0.y * w0.y + x0.z * w0.z + x0.w * w0.w
                + x1.x * w1.x + x1.y * w1.y + x1.z * w1.z + x1.w * w1.w;
        #pragma unroll
        for (int msk = 16; msk >= 1; msk >>= 1) d += __shfl_xor(d, msk, 32);
        if (lane == 0) {
            float l = d + bias[lab] - neg_log_expected_count(lab);
            acc += sigmoid_xent(l, tgt);
        }
    }
    if (lane == 0) redw[wv] = acc;
    __syncthreads();
    if (tid == 0) {
        float s = 0.f;
        #pragma unroll
        for (int i = 0; i < 8; ++i) s += redw[i];
        tpart[b] = s;
    }
}

// ---------------------------------------------------------------------------
// Final fixed-order reduction.
// ---------------------------------------------------------------------------
__global__ __launch_bounds__(256) void nce_reduce_kernel(
    const float* __restrict__ part, const float* __restrict__ tpart,
    float* __restrict__ out)
{
    int b = blockIdx.x * blockDim.x + threadIdx.x;
    if (b < BATCH) {
        float s = tpart[b];
        for (int c = 0; c < NCH; ++c) s += part[(size_t)b * NCH + c];
        out[b] = s;
    }
}

extern "C" void kernel_launch(void* const* d_in, const int* in_sizes, int n_in,
                              void* d_out, int out_size, void* d_ws, size_t ws_size,
                              hipStream_t stream) {
    const float* inp     = (const float*)d_in[0];   // [256, 256]
    const float* w       = (const float*)d_in[1];   // [100000, 256]
    const float* bias    = (const float*)d_in[2];   // [100000]
    const int*   labels  = (const int*)d_in[3];     // [256, 1024]
    const int*   sampled = (const int*)d_in[4];     // [16384]
    float*       out     = (float*)d_out;           // [256]

    float* part  = (float*)d_ws;                    // [256 * 64] sampled partials
    float* tpart = part + (size_t)BATCH * NCH;      // [256] true partials

    dim3 gridS(NCH, BATCH / MT);                    // 64 x 16 blocks
    nce_sampled_kernel<<<gridS, 256, 0, stream>>>(inp, w, bias, sampled, part);
    nce_true_kernel<<<BATCH, 256, 0, stream>>>(inp, w, bias, labels, tpart);
    nce_reduce_kernel<<<1, 256, 0, stream>>>(part, tpart, out);
}